// ParallelCellBatch_25185688224002
// MI455X (gfx1250) — compile-verified
//
#include <hip/hip_runtime.h>
#include <hip/hip_bf16.h>
#include <cstdint>
#include <cstddef>

#define N_NODES   4096
#define FEATD     128
#define NGRAPH    64
#define NODESPER  64
#define HIDD      512
#define NHEAD     8
#define DHD       64
#define NCELLS    64
#define LMAXD     512

typedef float v2f __attribute__((ext_vector_type(2)));
typedef float v8f __attribute__((ext_vector_type(8)));

__device__ __forceinline__ v8f wmma4(v2f a, v2f b, v8f c) {
  // D = A(16x4) * B(4x16) + C(16x16), fp32 WMMA
  return __builtin_amdgcn_wmma_f32_16x16x4_f32(false, a, false, b, (short)0, c, false, false);
}

// ---------------- stage 0: zero accumulators (deg + fixed-point agg) ----------------
__global__ void zero_u64_kernel(unsigned long long* p, long n) {
  long i = (long)blockIdx.x * blockDim.x + threadIdx.x;
  if (i < n) p[i] = 0ull;
}

// ---------------- stage 1: edge scatter (deterministic int64 fixed point) ----------------
__global__ void edge_kernel(const float* __restrict__ x, const int* __restrict__ edge,
                            int* __restrict__ deg, unsigned long long* __restrict__ agg, int E) {
  int e = blockIdx.x * blockDim.x + threadIdx.x;
  if (e >= E) return;
  int s = edge[e], d = edge[E + e];
  atomicAdd(&deg[d], 1);
  const float* xs = x + (size_t)s * FEATD;
  unsigned long long* ap = agg + (size_t)d * FEATD;
  for (int f = 0; f < FEATD; ++f) {
    long long q = (long long)llrintf(xs[f] * 4294967296.0f);
    atomicAdd(&ap[f], (unsigned long long)q);
  }
}

// ---------------- stage 2: xn = x + agg/deg, per-graph mean pool ----------------
__global__ void pool_kernel(const float* __restrict__ x, const int* __restrict__ deg,
                            const unsigned long long* __restrict__ agg,
                            float* __restrict__ pooled) {
  int b = blockIdx.x, f = threadIdx.x;  // 64 blocks x 128 threads
  float acc = 0.f;
  for (int i = 0; i < NODESPER; ++i) {
    int n = b * NODESPER + i;
    float dg = (float)max(deg[n], 1);
    float a = (float)(long long)agg[(size_t)n * FEATD + f] * (1.0f / 4294967296.0f);
    acc += x[(size_t)n * FEATD + f] + a / dg;
  }
  pooled[b * FEATD + f] = acc * (1.0f / NODESPER);
}

// ---------------- stage 3: per-gene GEMM + ReLU + ragged scatter into padded ----------------
__global__ void __launch_bounds__(256) gene_gemm_kernel(
    const float* __restrict__ pooled, const float* __restrict__ gene_W,
    const float* __restrict__ gene_b, const int* __restrict__ cell_ids,
    const int* __restrict__ rank, float* __restrict__ padded) {
  __shared__ float sA[64 * 132];  // pooled staged, padded stride (bank-conflict free)
  int tid = threadIdx.x;
  for (int idx = tid; idx < 64 * FEATD; idx += 256) {
    int r = idx >> 7, c = idx & 127;
    sA[r * 132 + c] = pooled[idx];
  }
  __syncthreads();
  int g = blockIdx.x;
  int wave = tid >> 5, lane = tid & 31;
  int col16 = lane & 15, khalf = lane >> 4;
  const float* Wg = gene_W + (size_t)g * FEATD * HIDD;
  float* dst = padded + ((size_t)cell_ids[g] * LMAXD + (size_t)rank[g] * NGRAPH) * HIDD;
  for (int strip = wave; strip < HIDD / 16; strip += 8) {  // 32 column strips
    int n0 = strip * 16;
    v8f acc[4] = {};
    for (int kk = 0; kk < FEATD; kk += 4) {
      int kr = kk + 2 * khalf;
      v2f bf;
      bf[0] = Wg[(size_t)kr * HIDD + n0 + col16];
      bf[1] = Wg[(size_t)(kr + 1) * HIDD + n0 + col16];
#pragma unroll
      for (int mt = 0; mt < 4; ++mt) {  // reuse B across all 4 M-tiles
        int row = mt * 16 + col16;
        v2f af;
        af[0] = sA[row * 132 + kr];
        af[1] = sA[row * 132 + kr + 1];
        acc[mt] = wmma4(af, bf, acc[mt]);
      }
    }
    float bias = gene_b[(size_t)g * HIDD + n0 + col16];
#pragma unroll
    for (int mt = 0; mt < 4; ++mt)
#pragma unroll
      for (int r = 0; r < 8; ++r) {
        int b = mt * 16 + r + 8 * khalf;  // batch index (row l = rank*64 + b)
        float v = acc[mt][r] + bias;
        dst[(size_t)b * HIDD + n0 + col16] = fmaxf(v, 0.f);
      }
  }
}

// ---------------- stage 4: Q/K/V projections (WMMA tiled) ----------------
__global__ void __launch_bounds__(256) qkv_kernel(
    const float* __restrict__ padded,
    const float* __restrict__ Wq, const float* __restrict__ bq,
    const float* __restrict__ Wk, const float* __restrict__ bk,
    const float* __restrict__ Wv, const float* __restrict__ bv,
    float* __restrict__ qo, float* __restrict__ ko, float* __restrict__ vo) {
  __shared__ float sA[64 * 17];
  int mb = blockIdx.x, which = blockIdx.y;
  const float* W = which == 0 ? Wq : (which == 1 ? Wk : Wv);
  const float* bias = which == 0 ? bq : (which == 1 ? bk : bv);
  float* out = which == 0 ? qo : (which == 1 ? ko : vo);
  int tid = threadIdx.x, wave = tid >> 5, lane = tid & 31;
  int col16 = lane & 15, khalf = lane >> 4;
  const float* Arows = padded + (size_t)mb * 64 * HIDD;
  v8f acc[16] = {};
  for (int kk0 = 0; kk0 < HIDD; kk0 += 16) {
    __syncthreads();
    for (int idx = tid; idx < 64 * 16; idx += 256) {
      int r = idx >> 4, c = idx & 15;
      sA[r * 17 + c] = Arows[(size_t)r * HIDD + kk0 + c];
    }
    __syncthreads();
    for (int k4 = 0; k4 < 16; k4 += 4) {
      int kr = k4 + 2 * khalf;
      v2f bf[4];
#pragma unroll
      for (int nt = 0; nt < 4; ++nt) {
        int n = wave * 64 + nt * 16 + col16;
        bf[nt][0] = W[(size_t)(kk0 + kr) * HIDD + n];
        bf[nt][1] = W[(size_t)(kk0 + kr + 1) * HIDD + n];
      }
#pragma unroll
      for (int mt = 0; mt < 4; ++mt) {
        v2f af;
        af[0] = sA[(mt * 16 + col16) * 17 + kr];
        af[1] = sA[(mt * 16 + col16) * 17 + kr + 1];
#pragma unroll
        for (int nt = 0; nt < 4; ++nt)
          acc[mt * 4 + nt] = wmma4(af, bf[nt], acc[mt * 4 + nt]);
      }
    }
  }
#pragma unroll
  for (int mt = 0; mt < 4; ++mt)
#pragma unroll
    for (int nt = 0; nt < 4; ++nt) {
      int n = wave * 64 + nt * 16 + col16;
      float bb = bias[n];
#pragma unroll
      for (int r = 0; r < 8; ++r) {
        int row = mb * 64 + mt * 16 + r + 8 * khalf;
        out[(size_t)row * HIDD + n] = acc[mt * 4 + nt][r] + bb;
      }
    }
}

// ---------------- stage 5: per (cell,head,qtile) attention, WMMA for S=QK^T and O=PV ----------------
__global__ void __launch_bounds__(32) attn_kernel(
    const float* __restrict__ qb, const float* __restrict__ kb,
    const float* __restrict__ vb, const int* __restrict__ counts,
    float* __restrict__ attnout) {
  __shared__ float sS[16 * 516];  // 16 query rows x up to 512 keys (padded stride)
  int qt = blockIdx.x, h = blockIdx.y, c = blockIdx.z;
  int seqlen = counts[c] * NGRAPH;          // multiple of 64
  if (qt * 16 >= seqlen) return;            // wave-uniform: padded queries skipped
  int lane = threadIdx.x, col16 = lane & 15, khalf = lane >> 4;
  const float* qp = qb + (size_t)c * LMAXD * HIDD + (size_t)h * DHD;
  const float* kp = kb + (size_t)c * LMAXD * HIDD + (size_t)h * DHD;
  const float* vp = vb + (size_t)c * LMAXD * HIDD + (size_t)h * DHD;
  // preload Q fragments for 16 K-steps covering dh=64
  v2f qa[16];
  {
    const float* qrow = qp + (size_t)(qt * 16 + col16) * HIDD;
#pragma unroll
    for (int i = 0; i < 16; ++i) {
      int d = i * 4 + 2 * khalf;
      qa[i][0] = qrow[d];
      qa[i][1] = qrow[d + 1];
    }
  }
  int nkt = seqlen / 16;  // only valid keys -> no masking needed
  for (int j = 0; j < nkt; ++j) {
    int kbase = j * 16;
    v8f s = {};
    const float* krow = kp + (size_t)(kbase + col16) * HIDD;
#pragma unroll
    for (int i = 0; i < 16; ++i) {
      int d = i * 4 + 2 * khalf;
      v2f bf;
      bf[0] = krow[d];
      bf[1] = krow[d + 1];
      s = wmma4(qa[i], bf, s);
    }
#pragma unroll
    for (int r = 0; r < 8; ++r)
      sS[(r + 8 * khalf) * 516 + kbase + col16] = s[r] * 0.125f;  // 1/sqrt(64)
  }
  __syncthreads();
  // softmax: 2 lanes per row, each scans half the keys
  {
    int row = lane & 15, half = lane >> 4;
    int c0 = half * (seqlen / 2), c1 = c0 + seqlen / 2;
    float* rp = sS + row * 516;
    float m = -3.4e38f;
    for (int i = c0; i < c1; ++i) m = fmaxf(m, rp[i]);
    m = fmaxf(m, __shfl_xor(m, 16, 32));
    float sum = 0.f;
    for (int i = c0; i < c1; ++i) { float e = expf(rp[i] - m); rp[i] = e; sum += e; }
    sum += __shfl_xor(sum, 16, 32);
    float inv = 1.f / sum;
    for (int i = c0; i < c1; ++i) rp[i] *= inv;
  }
  __syncthreads();
  // O(16x64) = P(16 x seqlen) @ V(seqlen x 64)
  v8f o[4] = {};
  for (int kk = 0; kk < seqlen; kk += 4) {
    int kr = kk + 2 * khalf;
    v2f af;
    af[0] = sS[col16 * 516 + kr];
    af[1] = sS[col16 * 516 + kr + 1];
#pragma unroll
    for (int nt = 0; nt < 4; ++nt) {
      v2f bf;
      bf[0] = vp[(size_t)kr * HIDD + nt * 16 + col16];
      bf[1] = vp[(size_t)(kr + 1) * HIDD + nt * 16 + col16];
      o[nt] = wmma4(af, bf, o[nt]);
    }
  }
  float* op = attnout + ((size_t)c * LMAXD + qt * 16) * HIDD + (size_t)h * DHD;
#pragma unroll
  for (int nt = 0; nt < 4; ++nt)
#pragma unroll
    for (int r = 0; r < 8; ++r)
      op[(size_t)(r + 8 * khalf) * HIDD + nt * 16 + col16] = o[nt][r];
}

// ---------------- stage 6: masked mean over valid rows (commutes with Wo) ----------------
__global__ void mean_kernel(const float* __restrict__ attnout, const int* __restrict__ counts,
                            float* __restrict__ xbar) {
  int c = blockIdx.x, tid = threadIdx.x;
  int seqlen = counts[c] * NGRAPH;
  float inv = 1.f / (float)seqlen;
  for (int hh = tid; hh < HIDD; hh += 256) {
    float acc = 0.f;
    const float* p = attnout + (size_t)c * LMAXD * HIDD + hh;
    for (int l = 0; l < seqlen; ++l) acc += p[(size_t)l * HIDD];
    xbar[c * HIDD + hh] = acc * inv;
  }
}

// ---------------- small dense layers (bandwidth-trivial) ----------------
__global__ void linear_kernel(const float* __restrict__ A, const float* __restrict__ W,
                              const float* __restrict__ bias, float* __restrict__ out,
                              int M, int K, int Nn, int relu) {
  int idx = blockIdx.x * blockDim.x + threadIdx.x;
  if (idx >= M * Nn) return;
  int m = idx / Nn, j = idx - m * Nn;
  const float* a = A + (size_t)m * K;
  float acc = bias[j];
  for (int k = 0; k < K; ++k) acc += a[k] * W[(size_t)k * Nn + j];
  out[idx] = relu ? fmaxf(acc, 0.f) : acc;
}

__global__ void ln_kernel(const float* __restrict__ h, const float* __restrict__ g,
                          const float* __restrict__ b, float* __restrict__ out) {
  __shared__ float red[256];
  int c = blockIdx.x, tid = threadIdx.x;
  const float* row = h + (size_t)c * HIDD;
  float s = 0.f;
  for (int j = tid; j < HIDD; j += 256) s += row[j];
  red[tid] = s; __syncthreads();
  for (int st = 128; st > 0; st >>= 1) { if (tid < st) red[tid] += red[tid + st]; __syncthreads(); }
  float mu = red[0] * (1.0f / HIDD);
  __syncthreads();
  float s2 = 0.f;
  for (int j = tid; j < HIDD; j += 256) { float d = row[j] - mu; s2 += d * d; }
  red[tid] = s2; __syncthreads();
  for (int st = 128; st > 0; st >>= 1) { if (tid < st) red[tid] += red[tid + st]; __syncthreads(); }
  float inv = rsqrtf(red[0] * (1.0f / HIDD) + 1e-5f);
  for (int j = tid; j < HIDD; j += 256)
    out[(size_t)c * HIDD + j] = (row[j] - mu) * inv * g[j] + b[j];
}

__global__ void aff_kernel(const float* __restrict__ integ, const float* __restrict__ Aw1,
                           const float* __restrict__ Ab1, const float* __restrict__ Aw2,
                           const float* __restrict__ Ab2, float* __restrict__ aff) {
  __shared__ float sI[HIDD];
  __shared__ float red[256];
  int c = blockIdx.x, tid = threadIdx.x;
  for (int j = tid; j < HIDD; j += 256) sI[j] = integ[(size_t)c * HIDD + j];
  __syncthreads();
  float part = 0.f;
  for (int kidx = tid; kidx < HIDD; kidx += 256) {
    float acc = Ab1[(size_t)c * HIDD + kidx];
    const float* Wc = Aw1 + (size_t)c * HIDD * HIDD + kidx;
    for (int hh = 0; hh < HIDD; ++hh) acc += sI[hh] * Wc[(size_t)hh * HIDD];
    part += fmaxf(acc, 0.f) * Aw2[(size_t)c * HIDD + kidx];
  }
  red[tid] = part; __syncthreads();
  for (int st = 128; st > 0; st >>= 1) { if (tid < st) red[tid] += red[tid + st]; __syncthreads(); }
  if (tid == 0) aff[c] = 1.f / (1.f + expf(-(red[0] + Ab2[c])));
}

extern "C" void kernel_launch(void* const* d_in, const int* in_sizes, int n_in,
                              void* d_out, int out_size, void* d_ws, size_t ws_size,
                              hipStream_t stream) {
  const float* x       = (const float*)d_in[0];
  const int*   edge    = (const int*)d_in[1];
  const float* gene_W  = (const float*)d_in[3];
  const float* gene_b  = (const float*)d_in[4];
  const int*   cell_ids= (const int*)d_in[5];
  const int*   rank    = (const int*)d_in[6];
  const int*   counts  = (const int*)d_in[7];
  const float* Wq = (const float*)d_in[8];   const float* bq = (const float*)d_in[9];
  const float* Wk = (const float*)d_in[10];  const float* bk = (const float*)d_in[11];
  const float* Wv = (const float*)d_in[12];  const float* bv = (const float*)d_in[13];
  const float* Wo = (const float*)d_in[14];  const float* bo = (const float*)d_in[15];
  const float* Wi1 = (const float*)d_in[16]; const float* bi1 = (const float*)d_in[17];
  const float* Wi2 = (const float*)d_in[18]; const float* bi2 = (const float*)d_in[19];
  const float* ln_g = (const float*)d_in[20]; const float* ln_b = (const float*)d_in[21];
  const float* Aw1 = (const float*)d_in[22]; const float* Ab1 = (const float*)d_in[23];
  const float* Aw2 = (const float*)d_in[24]; const float* Ab2 = (const float*)d_in[25];
  int G = in_sizes[5];          // 317 genes
  int E = in_sizes[1] / 2;      // 65536 edges
  float* out = (float*)d_out;   // [0,64): aff; [64, 64+32768): integrated

  // -------- workspace carve --------
  char* wsb = (char*)d_ws;
  size_t off = 0;
  int* deg = (int*)(wsb + off);                   off += (size_t)N_NODES * 4;
  unsigned long long* agg = (unsigned long long*)(wsb + off); off += (size_t)N_NODES * FEATD * 8;
  float* pooled = (float*)(wsb + off);            off += (size_t)NGRAPH * FEATD * 4;
  float* padded = (float*)(wsb + off);            off += (size_t)NCELLS * LMAXD * HIDD * 4;
  float* qbuf = (float*)(wsb + off);              off += (size_t)NCELLS * LMAXD * HIDD * 4;
  float* kbuf = (float*)(wsb + off);              off += (size_t)NCELLS * LMAXD * HIDD * 4;
  float* vbuf = (float*)(wsb + off);              off += (size_t)NCELLS * LMAXD * HIDD * 4;
  float* attnout = padded;  // padded is dead after qkv -> alias
  float* xbar  = (float*)(wsb + off);             off += (size_t)NCELLS * HIDD * 4;
  float* meanh = (float*)(wsb + off);             off += (size_t)NCELLS * HIDD * 4;
  float* h1    = (float*)(wsb + off);             off += (size_t)NCELLS * 2 * HIDD * 4;
  float* h2    = (float*)(wsb + off);             off += (size_t)NCELLS * HIDD * 4;

  // stage 0: zero deg + agg (contiguous)
  long nzero = ((size_t)N_NODES * 4 + (size_t)N_NODES * FEATD * 8) / 8;
  zero_u64_kernel<<<(int)((nzero + 255) / 256), 256, 0, stream>>>((unsigned long long*)wsb, nzero);
  // stage 1: edge scatter
  edge_kernel<<<(E + 255) / 256, 256, 0, stream>>>(x, edge, deg, agg, E);
  // stage 2: per-graph pooling
  pool_kernel<<<NGRAPH, FEATD, 0, stream>>>(x, deg, agg, pooled);
  // stage 3: gene GEMM + scatter into padded
  gene_gemm_kernel<<<G, 256, 0, stream>>>(pooled, gene_W, gene_b, cell_ids, rank, padded);
  // stage 4: Q/K/V projections
  qkv_kernel<<<dim3((NCELLS * LMAXD) / 64, 3), 256, 0, stream>>>(
      padded, Wq, bq, Wk, bk, Wv, bv, qbuf, kbuf, vbuf);
  // stage 5: attention (attnout aliases padded)
  attn_kernel<<<dim3(LMAXD / 16, NHEAD, NCELLS), 32, 0, stream>>>(
      qbuf, kbuf, vbuf, counts, attnout);
  // stage 6: masked mean over valid rows
  mean_kernel<<<NCELLS, 256, 0, stream>>>(attnout, counts, xbar);
  // stage 7: meanh = xbar @ Wo + bo (mean commutes with the linear layer)
  linear_kernel<<<(NCELLS * HIDD + 255) / 256, 256, 0, stream>>>(
      xbar, Wo, bo, meanh, NCELLS, HIDD, HIDD, 0);
  // stage 8: integrator MLP
  linear_kernel<<<(NCELLS * 2 * HIDD + 255) / 256, 256, 0, stream>>>(
      meanh, Wi1, bi1, h1, NCELLS, HIDD, 2 * HIDD, 1);
  linear_kernel<<<(NCELLS * HIDD + 255) / 256, 256, 0, stream>>>(
      h1, Wi2, bi2, h2, NCELLS, 2 * HIDD, HIDD, 0);
  // stage 9: LayerNorm -> integrated output
  ln_kernel<<<NCELLS, 256, 0, stream>>>(h2, ln_g, ln_b, out + NCELLS);
  // stage 10: per-cell affinity heads -> aff output
  aff_kernel<<<NCELLS, 256, 0, stream>>>(out + NCELLS, Aw1, Ab1, Aw2, Ab2, out);
}